// TroAttention_2284922602061
// MI455X (gfx1250) — compile-verified
//
#include <hip/hip_runtime.h>
#include <hip/hip_bf16.h>

typedef __attribute__((ext_vector_type(2))) float v2f;
typedef __attribute__((ext_vector_type(8))) float v8f;

#define LNUM 2
#define BNUM 32
#define TNUM 2048
#define HNUM 512

#if __has_builtin(__builtin_amdgcn_tanhf)
#define TANHF(x) __builtin_amdgcn_tanhf(x)
#else
#define TANHF(x) tanhf(x)
#endif

// ---------------------------------------------------------------------------
// Kernel 1: ha[B,H] = mean_L(hidden) @ Wh^T + bh, via V_WMMA_F32_16X16X4_F32.
// 64 waves total: 2 M-tiles (B=32 rows) x 32 N-tiles (H=512 cols).
// A-fragment layout (16x4 f32): lanes 0-15 row M=lane, VGPR0/1 = K{0,1};
//                               lanes 16-31 same rows,  VGPR0/1 = K{2,3}.
// B-fragment (4x16): lanes 0-15 col N=lane with K{0,1}; lanes 16-31 K{2,3}.
// C/D (16x16 f32): VGPR r = row m0+r (lanes 0-15) / m0+r+8 (lanes 16-31).
// ---------------------------------------------------------------------------
__global__ __launch_bounds__(256) void proj_wmma_kernel(
    const float* __restrict__ hidden, const float* __restrict__ Wh,
    const float* __restrict__ bh, float* __restrict__ ha)
{
  const int lane = threadIdx.x & 31;
  const int waveId = blockIdx.x * 8 + (threadIdx.x >> 5);  // 0..63
  const int m0 = (waveId & 1) * 16;
  const int n0 = (waveId >> 1) * 16;
  const int half = lane >> 4;
  const int lp = lane & 15;

  const float* __restrict__ h0 = hidden;              // layer 0 [B,H]
  const float* __restrict__ h1 = hidden + BNUM * HNUM; // layer 1

  const int arow = m0 + lp;   // A row supplied by this lane
  const int bcol = n0 + lp;   // output column == Wh row (B = Wh^T)

  v8f c;
#pragma unroll
  for (int i = 0; i < 8; ++i) c[i] = 0.0f;

  for (int k = 0; k < HNUM; k += 4) {
    const int ka = k + 2 * half;
    v2f e0 = *(const v2f*)(h0 + arow * HNUM + ka);
    v2f e1 = *(const v2f*)(h1 + arow * HNUM + ka);
    v2f a = (e0 + e1) * 0.5f;                         // mean over L=2
    v2f b = *(const v2f*)(Wh + bcol * HNUM + ka);     // Wh^T[ka..ka+1][bcol]
    c = __builtin_amdgcn_wmma_f32_16x16x4_f32(
        /*neg_a=*/false, a, /*neg_b=*/false, b,
        /*c_mod=*/(short)0, c, /*reuse_a=*/false, /*reuse_b=*/false);
  }

  const float bias = bh[bcol];
#pragma unroll
  for (int r = 0; r < 8; ++r) {
    const int row = m0 + r + 8 * half;
    ha[row * HNUM + bcol] = c[r] + bias;
  }
}

// ---------------------------------------------------------------------------
// Kernel 2: energy[b,t] = sum_h tanh(eo[t,b,h] + ha[b,h]) * Wo[h] + bo.
// One wave per (b,t). 512 floats per row, float4 per lane x 4 iters,
// fully coalesced 128B-per-lane-group streaming of the 134MB tensor.
// ---------------------------------------------------------------------------
__global__ __launch_bounds__(256) void energy_kernel(
    const float* __restrict__ eo, const float* __restrict__ ha,
    const float* __restrict__ Wo, const float* __restrict__ bo,
    float* __restrict__ energy)
{
  const int lane = threadIdx.x & 31;
  const int waveId = blockIdx.x * 8 + (threadIdx.x >> 5);  // 0..B*T-1
  const int b = waveId & (BNUM - 1);
  const int t = waveId >> 5;                               // / BNUM

  const float4* __restrict__ e4 =
      (const float4*)(eo + (size_t)(t * BNUM + b) * HNUM);
  const float4* __restrict__ a4 = (const float4*)(ha + b * HNUM);
  const float4* __restrict__ w4 = (const float4*)Wo;

  float acc = 0.0f;
#pragma unroll
  for (int i = 0; i < 4; ++i) {
    const int idx = lane + 32 * i;
    float4 e = e4[idx];
    float4 a = a4[idx];
    float4 w = w4[idx];
    acc += TANHF(e.x + a.x) * w.x;
    acc += TANHF(e.y + a.y) * w.y;
    acc += TANHF(e.z + a.z) * w.z;
    acc += TANHF(e.w + a.w) * w.w;
  }
#pragma unroll
  for (int off = 16; off; off >>= 1)
    acc += __shfl_xor(acc, off, 32);
  if (lane == 0) energy[b * TNUM + t] = acc + bo[0];
}

// ---------------------------------------------------------------------------
// Kernel 3: ragged masked softmax over T per batch row. One block per b.
// ---------------------------------------------------------------------------
__global__ __launch_bounds__(256) void softmax_kernel(
    const float* __restrict__ energy, const int* __restrict__ enc_len,
    float* __restrict__ out)
{
  __shared__ float red[256];
  const int b = blockIdx.x;
  const int tid = threadIdx.x;
  const int len = enc_len[b];
  const float* __restrict__ e = energy + b * TNUM;

  float vals[8];
  float mx = -3.0e38f;
#pragma unroll
  for (int j = 0; j < 8; ++j) {
    const int idx = tid + 256 * j;
    const float v = (idx < len) ? e[idx] : -3.0e38f;
    vals[j] = v;
    mx = fmaxf(mx, v);
  }
  red[tid] = mx;
  __syncthreads();
  for (int s = 128; s; s >>= 1) {
    if (tid < s) red[tid] = fmaxf(red[tid], red[tid + s]);
    __syncthreads();
  }
  mx = red[0];
  __syncthreads();

  float ex[8];
  float sum = 0.0f;
#pragma unroll
  for (int j = 0; j < 8; ++j) {
    const int idx = tid + 256 * j;
    const float v = (idx < len) ? __expf(vals[j] - mx) : 0.0f;
    ex[j] = v;
    sum += v;
  }
  red[tid] = sum;
  __syncthreads();
  for (int s = 128; s; s >>= 1) {
    if (tid < s) red[tid] += red[tid + s];
    __syncthreads();
  }
  const float inv = 1.0f / red[0];
#pragma unroll
  for (int j = 0; j < 8; ++j) {
    const int idx = tid + 256 * j;
    out[b * TNUM + idx] = ex[j] * inv;   // masked lanes have ex==0 -> 0
  }
}

extern "C" void kernel_launch(void* const* d_in, const int* in_sizes, int n_in,
                              void* d_out, int out_size, void* d_ws, size_t ws_size,
                              hipStream_t stream) {
  (void)in_sizes; (void)n_in; (void)out_size; (void)ws_size;
  const float* hidden  = (const float*)d_in[0];   // [L,B,H]
  const float* eo      = (const float*)d_in[1];   // [T,B,H]
  const int*   enc_len = (const int*)d_in[2];     // [B]
  const float* Wh      = (const float*)d_in[3];   // [H,H]
  const float* bh      = (const float*)d_in[4];   // [H]
  const float* Wo      = (const float*)d_in[5];   // [1,H]
  const float* bo      = (const float*)d_in[6];   // [1]
  float* out = (float*)d_out;                     // [B,T,1]

  float* ha     = (float*)d_ws;                   // [B,H]   64KB
  float* energy = ha + BNUM * HNUM;               // [B,T]  256KB

  hipLaunchKernelGGL(proj_wmma_kernel, dim3(8), dim3(256), 0, stream,
                     hidden, Wh, bh, ha);
  hipLaunchKernelGGL(energy_kernel, dim3((BNUM * TNUM) / 8), dim3(256), 0, stream,
                     eo, ha, Wo, bo, energy);
  hipLaunchKernelGGL(softmax_kernel, dim3(BNUM), dim3(256), 0, stream,
                     energy, enc_len, out);
}